// GAT_84782654423713
// MI455X (gfx1250) — compile-verified
//
#include <hip/hip_runtime.h>
#include <math.h>

#define IN_CH 128
#define HID 8
#define HEADS 8
#define F1 64            // HEADS*HID
#define OUT_CH 40
#define NEG_SLOPE 0.2f

typedef __attribute__((ext_vector_type(2))) float v2f;
typedef __attribute__((ext_vector_type(8))) float v8f;

// ---- order-preserving float<->uint encoding for atomic segment-max ----
// enc is strictly monotone; enc(any finite float) > 0, so memset(0) == -inf.
__device__ __forceinline__ unsigned fenc(float f) {
  unsigned u = __float_as_uint(f);
  return (u & 0x80000000u) ? ~u : (u | 0x80000000u);
}
__device__ __forceinline__ float fdec(unsigned e) {
  return (e & 0x80000000u) ? __uint_as_float(e & 0x7FFFFFFFu)
                           : __uint_as_float(~e);
}
__device__ __forceinline__ float lrelu(float x) { return x > 0.f ? x : NEG_SLOPE * x; }

// ======================= GEMM 1: H1[N,64] = X[N,128] @ W1^T =======================
// W1 is [64,128] row-major. One wave -> one 16x16 tile, K stepped by 4 (f32 WMMA).
__global__ void gat_gemm1(const float* __restrict__ X, const float* __restrict__ W,
                          float* __restrict__ H) {
  const int rowTile = blockIdx.x * 16;
  const int wave = threadIdx.x >> 5;          // 4 waves -> 4 col tiles (64 cols)
  const int lane = threadIdx.x & 31;
  const int colBase = wave * 16;
  const int nn = lane & 15;                   // M index (A) / N index (B,C)
  const int half = lane >> 4;                 // 0: K=0,1 / M=0..7 ; 1: K=2,3 / M=8..15
  const float* xrow = X + (size_t)(rowTile + nn) * IN_CH;
  const float* wrow = W + (size_t)(colBase + nn) * IN_CH;
  v8f acc = {};
#pragma unroll 4
  for (int k = 0; k < IN_CH; k += 4) {
    v2f a, b;
    a[0] = xrow[k + 2 * half];
    a[1] = xrow[k + 2 * half + 1];
    b[0] = wrow[k + 2 * half];
    b[1] = wrow[k + 2 * half + 1];
    acc = __builtin_amdgcn_wmma_f32_16x16x4_f32(false, a, false, b, (short)0, acc,
                                                false, false);
  }
  const int m0 = half * 8;
#pragma unroll
  for (int j = 0; j < 8; ++j)
    H[(size_t)(rowTile + m0 + j) * F1 + colBase + nn] = acc[j];
}

// ======================= GEMM 2: HH[N,40] = H2[N,64] @ W2^T =======================
// W2 is [40,64]. 3 col tiles (48), cols >= 40 masked (clamped loads + select).
__global__ void gat_gemm2(const float* __restrict__ H2, const float* __restrict__ W,
                          float* __restrict__ HH) {
  const int rowTile = blockIdx.x * 16;
  const int colBase = blockIdx.y * 16;
  const int lane = threadIdx.x & 31;
  const int nn = lane & 15;
  const int half = lane >> 4;
  const int col = colBase + nn;
  const int colc = col < OUT_CH ? col : OUT_CH - 1;
  const bool cok = col < OUT_CH;
  const float* hrow = H2 + (size_t)(rowTile + nn) * F1;
  const float* wrow = W + (size_t)colc * F1;
  v8f acc = {};
#pragma unroll 4
  for (int k = 0; k < F1; k += 4) {
    v2f a, b;
    a[0] = hrow[k + 2 * half];
    a[1] = hrow[k + 2 * half + 1];
    float b0 = wrow[k + 2 * half];
    float b1 = wrow[k + 2 * half + 1];
    b[0] = cok ? b0 : 0.f;
    b[1] = cok ? b1 : 0.f;
    acc = __builtin_amdgcn_wmma_f32_16x16x4_f32(false, a, false, b, (short)0, acc,
                                                false, false);
  }
  if (cok) {
    const int m0 = half * 8;
#pragma unroll
    for (int j = 0; j < 8; ++j)
      HH[(size_t)(rowTile + m0 + j) * OUT_CH + col] = acc[j];
  }
}

// =================== attention logits per node ===================
__global__ void gat_attn1(const float* __restrict__ H1, const float* __restrict__ asrc,
                          const float* __restrict__ adst, float* __restrict__ as1,
                          float* __restrict__ ad1, int n) {
  int t = blockIdx.x * blockDim.x + threadIdx.x;
  if (t >= n * HEADS) return;
  int node = t >> 3, h = t & 7;
  const float* hp = H1 + (size_t)node * F1 + h * HID;
  const float* s = asrc + h * HID;
  const float* d = adst + h * HID;
  float vs = 0.f, vd = 0.f;
#pragma unroll
  for (int c = 0; c < HID; ++c) { float x = hp[c]; vs += x * s[c]; vd += x * d[c]; }
  as1[t] = vs; ad1[t] = vd;
}

__global__ void gat_attn2(const float* __restrict__ HH, const float* __restrict__ asrc,
                          const float* __restrict__ adst, float* __restrict__ as2,
                          float* __restrict__ ad2, int n) {
  int t = blockIdx.x * blockDim.x + threadIdx.x;
  if (t >= n) return;
  const float* hp = HH + (size_t)t * OUT_CH;
  float vs = 0.f, vd = 0.f;
#pragma unroll
  for (int c = 0; c < OUT_CH; ++c) { float x = hp[c]; vs += x * asrc[c]; vd += x * adst[c]; }
  as2[t] = vs; ad2[t] = vd;
}

// =================== layer-1 edge passes (per edge x head) ===================
__global__ void gat_edge_max1(const int* __restrict__ src, const int* __restrict__ dst,
                              const float* __restrict__ as1, const float* __restrict__ ad1,
                              unsigned* __restrict__ emax, int eall) {
  int t = blockIdx.x * blockDim.x + threadIdx.x;
  if (t >= eall * HEADS) return;
  int e = t >> 3, h = t & 7;
  int s = src[e], d = dst[e];
  float v = lrelu(as1[s * HEADS + h] + ad1[d * HEADS + h]);
  atomicMax(&emax[d * HEADS + h], fenc(v));
}

__global__ void gat_edge_sum1(const int* __restrict__ src, const int* __restrict__ dst,
                              const float* __restrict__ as1, const float* __restrict__ ad1,
                              const unsigned* __restrict__ emax, float* __restrict__ den,
                              int eall) {
  int t = blockIdx.x * blockDim.x + threadIdx.x;
  if (t >= eall * HEADS) return;
  int e = t >> 3, h = t & 7;
  int s = src[e], d = dst[e];
  float v = lrelu(as1[s * HEADS + h] + ad1[d * HEADS + h]);
  float ex = expf(v - fdec(emax[d * HEADS + h]));
  atomicAdd(&den[d * HEADS + h], ex);
}

__global__ void gat_edge_scatter1(const int* __restrict__ src, const int* __restrict__ dst,
                                  const float* __restrict__ as1, const float* __restrict__ ad1,
                                  const unsigned* __restrict__ emax, const float* __restrict__ den,
                                  const float* __restrict__ H1, float* __restrict__ acc,
                                  int eall) {
  int t = blockIdx.x * blockDim.x + threadIdx.x;
  if (t >= eall * HEADS) return;
  int e = t >> 3, h = t & 7;
  int s = src[e], d = dst[e];
  const float* hp = H1 + (size_t)s * F1 + h * HID;
  __builtin_prefetch(hp, 0, 0);  // global_prefetch_b8 for the gather row
  float v = lrelu(as1[s * HEADS + h] + ad1[d * HEADS + h]);
  float alpha = expf(v - fdec(emax[d * HEADS + h])) / den[d * HEADS + h];
  float* ap = acc + (size_t)d * F1 + h * HID;
#pragma unroll
  for (int c = 0; c < HID; ++c) atomicAdd(&ap[c], hp[c] * alpha);
}

// =================== layer-2 edge passes (1 head, per edge) ===================
__global__ void gat_edge_max2(const int* __restrict__ src, const int* __restrict__ dst,
                              const float* __restrict__ as2, const float* __restrict__ ad2,
                              unsigned* __restrict__ emax, int eall) {
  int e = blockIdx.x * blockDim.x + threadIdx.x;
  if (e >= eall) return;
  float v = lrelu(as2[src[e]] + ad2[dst[e]]);
  atomicMax(&emax[dst[e]], fenc(v));
}

__global__ void gat_edge_sum2(const int* __restrict__ src, const int* __restrict__ dst,
                              const float* __restrict__ as2, const float* __restrict__ ad2,
                              const unsigned* __restrict__ emax, float* __restrict__ den,
                              int eall) {
  int e = blockIdx.x * blockDim.x + threadIdx.x;
  if (e >= eall) return;
  int d = dst[e];
  float v = lrelu(as2[src[e]] + ad2[d]);
  atomicAdd(&den[d], expf(v - fdec(emax[d])));
}

__global__ void gat_edge_scatter2(const int* __restrict__ src, const int* __restrict__ dst,
                                  const float* __restrict__ as2, const float* __restrict__ ad2,
                                  const unsigned* __restrict__ emax, const float* __restrict__ den,
                                  const float* __restrict__ HH, float* __restrict__ out,
                                  int eall) {
  int e = blockIdx.x * blockDim.x + threadIdx.x;
  if (e >= eall) return;
  int s = src[e], d = dst[e];
  const float* hp = HH + (size_t)s * OUT_CH;
  __builtin_prefetch(hp, 0, 0);
  float v = lrelu(as2[s] + ad2[d]);
  float alpha = expf(v - fdec(emax[d])) / den[d];
  float* op = out + (size_t)d * OUT_CH;
#pragma unroll
  for (int c = 0; c < OUT_CH; ++c) atomicAdd(&op[c], hp[c] * alpha);
}

// =================== finalize ===================
__global__ void gat_finalize1(float* __restrict__ acc, const float* __restrict__ b1, int n) {
  int t = blockIdx.x * blockDim.x + threadIdx.x;
  if (t >= n * F1) return;
  float x = acc[t] + b1[t & (F1 - 1)];
  acc[t] = x > 0.f ? x : expm1f(x);   // ELU, in place: acc becomes layer-2 input
}

__global__ void gat_finalize2(float* __restrict__ out, const float* __restrict__ b2, int n) {
  int t = blockIdx.x * blockDim.x + threadIdx.x;
  if (t >= n * OUT_CH) return;
  out[t] += b2[t % OUT_CH];
}

extern "C" void kernel_launch(void* const* d_in, const int* in_sizes, int n_in,
                              void* d_out, int out_size, void* d_ws, size_t ws_size,
                              hipStream_t stream) {
  const float* x     = (const float*)d_in[0];
  const int*   ei    = (const int*)d_in[1];
  const float* W1    = (const float*)d_in[2];
  const float* asrc1 = (const float*)d_in[3];
  const float* adst1 = (const float*)d_in[4];
  const float* b1    = (const float*)d_in[5];
  const float* W2    = (const float*)d_in[6];
  const float* asrc2 = (const float*)d_in[7];
  const float* adst2 = (const float*)d_in[8];
  const float* b2    = (const float*)d_in[9];
  float* out = (float*)d_out;

  const int n    = in_sizes[0] / IN_CH;   // 100000
  const int eall = in_sizes[1] / 2;       // 1700000
  const int* src = ei;
  const int* dst = ei + eall;

  // ---- carve workspace (256B-aligned regions) ----
  char* w = (char*)d_ws;
  auto carve = [&](size_t bytes) -> void* {
    void* p = (void*)w;
    w += (bytes + 255) & ~(size_t)255;
    return p;
  };
  float*    H1    = (float*)carve((size_t)n * F1 * 4);
  float*    as1   = (float*)carve((size_t)n * HEADS * 4);
  float*    ad1   = (float*)carve((size_t)n * HEADS * 4);
  unsigned* emax1 = (unsigned*)carve((size_t)n * HEADS * 4);
  float*    den1  = (float*)carve((size_t)n * HEADS * 4);
  float*    acc1  = (float*)carve((size_t)n * F1 * 4);     // becomes H2 after ELU
  float*    HH    = (float*)carve((size_t)n * OUT_CH * 4);
  float*    as2   = (float*)carve((size_t)n * 4);
  float*    ad2   = (float*)carve((size_t)n * 4);
  unsigned* emax2 = (unsigned*)carve((size_t)n * 4);
  float*    den2  = (float*)carve((size_t)n * 4);

  hipMemsetAsync(emax1, 0, (size_t)n * HEADS * 4, stream);  // enc(-inf) == 0
  hipMemsetAsync(den1,  0, (size_t)n * HEADS * 4, stream);
  hipMemsetAsync(acc1,  0, (size_t)n * F1 * 4, stream);
  hipMemsetAsync(emax2, 0, (size_t)n * 4, stream);
  hipMemsetAsync(den2,  0, (size_t)n * 4, stream);
  hipMemsetAsync(out,   0, (size_t)n * OUT_CH * 4, stream);

  const int tiles = n / 16;                 // N divisible by 16
  const int B = 256;
  const int nh = n * HEADS, eh = eall * HEADS;

  // ---- layer 1 ----
  gat_gemm1<<<tiles, 128, 0, stream>>>(x, W1, H1);
  gat_attn1<<<(nh + B - 1) / B, B, 0, stream>>>(H1, asrc1, adst1, as1, ad1, n);
  gat_edge_max1<<<(eh + B - 1) / B, B, 0, stream>>>(src, dst, as1, ad1, emax1, eall);
  gat_edge_sum1<<<(eh + B - 1) / B, B, 0, stream>>>(src, dst, as1, ad1, emax1, den1, eall);
  gat_edge_scatter1<<<(eh + B - 1) / B, B, 0, stream>>>(src, dst, as1, ad1, emax1, den1,
                                                        H1, acc1, eall);
  gat_finalize1<<<(n * F1 + B - 1) / B, B, 0, stream>>>(acc1, b1, n);

  // ---- layer 2 ----
  gat_gemm2<<<dim3(tiles, 3), 32, 0, stream>>>(acc1, W2, HH);
  gat_attn2<<<(n + B - 1) / B, B, 0, stream>>>(HH, asrc2, adst2, as2, ad2, n);
  gat_edge_max2<<<(eall + B - 1) / B, B, 0, stream>>>(src, dst, as2, ad2, emax2, eall);
  gat_edge_sum2<<<(eall + B - 1) / B, B, 0, stream>>>(src, dst, as2, ad2, emax2, den2, eall);
  gat_edge_scatter2<<<(eall + B - 1) / B, B, 0, stream>>>(src, dst, as2, ad2, emax2, den2,
                                                          HH, out, eall);
  gat_finalize2<<<(n * OUT_CH + B - 1) / B, B, 0, stream>>>(out, b2, n);
}